// Generator_12902081757484
// MI455X (gfx1250) — compile-verified
//
#include <hip/hip_runtime.h>
#include <hip/hip_bf16.h>

// ---------------------------------------------------------------------------
// MPS perfect-sampling pipeline for MI455X (gfx1250, wave32, WMMA f32 16x16x4,
// TDM tensor_load_to_lds staging for the sequential environment scan).
// ---------------------------------------------------------------------------

#define NSITES   1024
#define CHI      128
#define TWO_CHI  256
#define AUG_COLS 384   // [ (I+A) 256 cols | RHS 128 cols ]
#define NB       16
#define CHUNK    256   // sites factorized per launch (workspace reuse)

typedef __attribute__((ext_vector_type(2))) float v2f;
typedef __attribute__((ext_vector_type(8))) float v8f;
typedef __attribute__((ext_vector_type(4))) unsigned int u32x4;
typedef __attribute__((ext_vector_type(4))) int i32x4;
typedef __attribute__((ext_vector_type(8))) int i32x8;

__device__ __forceinline__ v8f wmma4(v2f a, v2f b, v8f c) {
  // V_WMMA_F32_16X16X4_F32 : D = A(16x4) * B(4x16) + C(16x16), all f32
  return __builtin_amdgcn_wmma_f32_16x16x4_f32(false, a, false, b, (short)0, c,
                                               false, false);
}

#define ZERO8 v8f{0.f, 0.f, 0.f, 0.f, 0.f, 0.f, 0.f, 0.f}

#if defined(__has_builtin)
#if __has_builtin(__builtin_amdgcn_tensor_load_to_lds) && \
    __has_builtin(__builtin_amdgcn_s_wait_tensorcnt)
#define HAVE_TDM 1
#endif
#endif
#ifndef HAVE_TDM
#define HAVE_TDM 0
#endif

#if HAVE_TDM
// Issue a TDM load of a 2D f32 tile (rows x cols, row stride stride0 elements)
// from global memory into LDS at byte offset lds_off. D# packing follows
// CDNA5 ISA (2D tensor: groups 2/3 zero).
__device__ __forceinline__ void tdm_load_2d_f32(unsigned lds_off,
                                                const float* gptr, int cols,
                                                int rows, int stride0) {
  const unsigned long long ga = (unsigned long long)(uintptr_t)gptr;
  u32x4 g0;
  g0[0] = 1u;                                   // count=1, user mode, no gather
  g0[1] = lds_off;                              // lds_addr (bytes)
  g0[2] = (unsigned)(ga & 0xFFFFFFFFu);         // global_addr[31:0]
  g0[3] = (unsigned)((ga >> 32) & 0x1FFFFFFu)   // global_addr[56:32]
          | (2u << 30);                         // type = 2 ("image")
  i32x8 g1;
  g1[0] = (2 << 16);                            // data_size=4B, mask=0
  g1[1] = (cols & 0xFFFF) << 16;                // tensor_dim0[15:0]  @ bits 63:48
  g1[2] = ((cols >> 16) & 0xFFFF)               // tensor_dim0[31:16] @ bits 79:64
          | ((rows & 0xFFFF) << 16);            // tensor_dim1[15:0]  @ bits 95:80
  g1[3] = ((rows >> 16) & 0xFFFF)               // tensor_dim1[31:16] @ bits 111:96
          | ((cols & 0xFFFF) << 16);            // tile_dim0          @ bits 127:112
  g1[4] = (rows & 0xFFFF);                      // tile_dim1 (tile_dim2 = 0)
  g1[5] = stride0;                              // tensor_dim0_stride[31:0]
  g1[6] = 0;                                    // stride0 hi | dim1_stride lo
  g1[7] = 0;
  i32x4 z4 = {0, 0, 0, 0};
#if __clang_major__ >= 23
  i32x8 z8 = {0, 0, 0, 0, 0, 0, 0, 0};
  __builtin_amdgcn_tensor_load_to_lds(g0, g1, z4, z4, z8, 0);
#else
  __builtin_amdgcn_tensor_load_to_lds(g0, g1, z4, z4, 0);
#endif
}
#endif

// workspace layout (in floats)
constexpr size_t SZ_T  = (size_t)NSITES * TWO_CHI * CHI;      // isometries W
constexpr size_t SZ_R  = (size_t)(NSITES + 1) * CHI * CHI;    // right envs
constexpr size_t SZ_S  = (size_t)CHI * CHI                    // Lenv
                       + 2 * (size_t)TWO_CHI * CHI            // Q1 / M3 (Q1 doubles as H)
                       + (size_t)CHI * CHI;                   // P1
constexpr size_t OFF_T  = 0;
constexpr size_t OFF_R  = OFF_T + SZ_T;
constexpr size_t OFF_S  = OFF_R + SZ_R;
constexpr size_t OFF_LU = OFF_S + SZ_S;                       // CHUNK*256*384

// ---------------------------------------------------------------------------
// Kernel 1: build augmented system  [ I+A | (I-A)[:, :CHI] ]  per site
//   A = 0.5*(S - S^T),  S = [X | 0] (256x256), X = params[site] (256x128)
// ---------------------------------------------------------------------------
__global__ void build_aug_kernel(const float* __restrict__ params,
                                 float* __restrict__ aug, int site0) {
  const int site = site0 + blockIdx.x;
  const float* X = params + (size_t)site * TWO_CHI * CHI;
  float* A = aug + (size_t)blockIdx.x * TWO_CHI * AUG_COLS;
  const int i = threadIdx.x;  // row 0..255
  for (int j = 0; j < TWO_CHI; ++j) {
    const float xij = (j < CHI) ? X[(size_t)i * CHI + j] : 0.0f;
    const float xji = (i < CHI) ? X[(size_t)j * CHI + i] : 0.0f;
    const float a = 0.5f * (xij - xji);
    A[(size_t)i * AUG_COLS + j] = a + ((i == j) ? 1.0f : 0.0f);
    if (j < CHI)
      A[(size_t)i * AUG_COLS + TWO_CHI + j] = ((i == j) ? 1.0f : 0.0f) - a;
  }
}

// ---------------------------------------------------------------------------
// Kernel 2: per-site blocked LU (no pivot) + back substitution.
// One workgroup per site, 256 threads (8 wave32). Trailing updates and the
// back-substitution GEMMs run on V_WMMA_F32_16X16X4_F32.
// Writes W = Q[:, :CHI] (256x128 row-major) into Tbuf[site].
// ---------------------------------------------------------------------------
__global__ void lu_solve_kernel(float* __restrict__ aug,
                                float* __restrict__ Tbuf, int site0) {
  __shared__ float panel[TWO_CHI][NB + 1];          // current column panel (L)
  __shared__ float urow[NB][AUG_COLS - NB + 4];     // current U row-block
  __shared__ float wblk[NB][CHI];                   // back-subst block

  float* A = aug + (size_t)blockIdx.x * TWO_CHI * AUG_COLS;
  float* W = Tbuf + (size_t)(site0 + blockIdx.x) * TWO_CHI * CHI;

  const int tid = threadIdx.x;
  const int lane = tid & 31, wv = tid >> 5;
  const int ln16 = lane & 15, hi = lane >> 4;

  // ---------------- right-looking blocked LU on [B0 | RHS] ----------------
  for (int kb = 0; kb < TWO_CHI; kb += NB) {
    const int nr = TWO_CHI - kb;
    // load panel (rows kb..255, cols kb..kb+15)
    for (int idx = tid; idx < nr * NB; idx += 256) {
      const int r = idx >> 4, c = idx & 15;
      panel[r][c] = A[(size_t)(kb + r) * AUG_COLS + kb + c];
    }
    __syncthreads();
    // unblocked panel factorization
    for (int j = 0; j < NB; ++j) {
      const float inv = 1.0f / panel[j][j];
      for (int r = j + 1 + tid; r < nr; r += 256) {
        const float l = panel[r][j] * inv;
        panel[r][j] = l;
        for (int jj = j + 1; jj < NB; ++jj) panel[r][jj] -= l * panel[j][jj];
      }
      __syncthreads();
    }
    // write panel back
    for (int idx = tid; idx < nr * NB; idx += 256) {
      const int r = idx >> 4, c = idx & 15;
      A[(size_t)(kb + r) * AUG_COLS + kb + c] = panel[r][c];
    }
    const int c0 = kb + NB;
    const int nc = AUG_COLS - c0;           // multiple of 16
    // load U row-block
    for (int idx = tid; idx < NB * nc; idx += 256) {
      const int r = idx / nc, c = idx % nc;
      urow[r][c] = A[(size_t)(kb + r) * AUG_COLS + c0 + c];
    }
    __syncthreads();
    // U12 <- L11^{-1} * U12 (unit-lower triangular solve)
    for (int j = 0; j < NB - 1; ++j) {
      for (int idx = tid; idx < (NB - 1 - j) * nc; idx += 256) {
        const int i2 = j + 1 + idx / nc, c = idx % nc;
        urow[i2][c] -= panel[i2][j] * urow[j][c];
      }
      __syncthreads();
    }
    // write U rows back
    for (int idx = tid; idx < NB * nc; idx += 256) {
      const int r = idx / nc, c = idx % nc;
      A[(size_t)(kb + r) * AUG_COLS + c0 + c] = urow[r][c];
    }
    // trailing update A22 -= L21 * U12 via WMMA (16x16 tiles, K=16)
    const int m = nr - NB;
    if (m > 0) {
      const int ntm = m >> 4, ntn = nc >> 4;
      const int tiles = ntm * ntn;
      for (int t = wv; t < tiles; t += 8) {
        const int tm = t / ntn, tn = t % ntn;
        v8f acc = ZERO8;
        const int ar = NB + tm * NB + ln16;   // local panel row of this lane
        for (int k0 = 0; k0 < NB; k0 += 4) {
          const int ka = k0 + 2 * hi;
          v2f a, b;
          a.x = panel[ar][ka];
          a.y = panel[ar][ka + 1];
          b.x = urow[ka][tn * NB + ln16];
          b.y = urow[ka + 1][tn * NB + ln16];
          acc = wmma4(a, b, acc);
        }
        float* base = A + (size_t)(kb + NB + tm * NB) * AUG_COLS + c0 + tn * NB;
        for (int r = 0; r < 8; ++r)
          base[(size_t)(r + 8 * hi) * AUG_COLS + ln16] -= acc[r];
      }
    }
    __threadfence();
    __syncthreads();
  }

  // ---------------- blocked back substitution: U * W = Y ----------------
  for (int kb = TWO_CHI - NB; kb >= 0; kb -= NB) {
    for (int idx = tid; idx < NB * NB; idx += 256) {
      const int r = idx >> 4, c = idx & 15;
      panel[r][c] = A[(size_t)(kb + r) * AUG_COLS + kb + c];
    }
    for (int idx = tid; idx < NB * CHI; idx += 256) {
      const int r = idx >> 7, c = idx & 127;
      wblk[r][c] = A[(size_t)(kb + r) * AUG_COLS + TWO_CHI + c];
    }
    __syncthreads();
    // 16x16 triangular solve against 128 RHS columns
    for (int i2 = NB - 1; i2 >= 0; --i2) {
      if (tid < CHI) {
        float t = wblk[i2][tid];
        for (int j = i2 + 1; j < NB; ++j) t -= panel[i2][j] * wblk[j][tid];
        wblk[i2][tid] = t / panel[i2][i2];
      }
      __syncthreads();
    }
    // emit solved rows of W
    for (int idx = tid; idx < NB * CHI; idx += 256) {
      const int r = idx >> 7, c = idx & 127;
      W[(size_t)(kb + r) * CHI + c] = wblk[r][c];
    }
    // update Y rows above: Y[0:kb,:] -= U[0:kb, kb:kb+16] * W16 (WMMA)
    if (kb > 0) {
      const int ntm = kb >> 4;
      const int tiles = ntm * (CHI / NB);
      for (int t = wv; t < tiles; t += 8) {
        const int tm = t >> 3, tn = t & 7;
        v8f acc = ZERO8;
        for (int k0 = 0; k0 < NB; k0 += 4) {
          const int ka = k0 + 2 * hi;
          v2f a, b;
          a.x = A[(size_t)(tm * NB + ln16) * AUG_COLS + kb + ka];
          a.y = A[(size_t)(tm * NB + ln16) * AUG_COLS + kb + ka + 1];
          b.x = wblk[ka][tn * NB + ln16];
          b.y = wblk[ka + 1][tn * NB + ln16];
          acc = wmma4(a, b, acc);
        }
        float* base = A + (size_t)(tm * NB) * AUG_COLS + TWO_CHI + tn * NB;
        for (int r = 0; r < 8; ++r)
          base[(size_t)(r + 8 * hi) * AUG_COLS + ln16] -= acc[r];
      }
    }
    __threadfence();
    __syncthreads();
  }
}

// ---------------------------------------------------------------------------
// Kernel 3: sequential right-environment scan (single persistent workgroup).
//   R_i[a][c] = sum_{s,b,d} T[a,s,b] T[c,s,d] R_{i+1}[b,d]
// The 128x128 R matrix is staged into LDS each step via the Tensor Data
// Mover (tensor_load_to_lds + s_wait_tensorcnt), falling back to a
// cooperative copy when the TDM builtin is unavailable.
// ---------------------------------------------------------------------------
__global__ void right_env_kernel(const float* __restrict__ Tbuf,
                                 float* __restrict__ Renv,
                                 float* __restrict__ Hbuf) {
  __shared__ float Rlds[CHI * CHI];   // 64 KB staging for R_{i+1}

  const int tid = threadIdx.x;
  const int lane = tid & 31, wv = tid >> 5;
  const int ln16 = lane & 15, hi = lane >> 4;

  // Renv[NSITES] = outer(r, r) = e0 e0^T
  float* Rlast = Renv + (size_t)NSITES * CHI * CHI;
  for (int idx = tid; idx < CHI * CHI; idx += 256)
    Rlast[idx] = (idx == 0) ? 1.0f : 0.0f;
  __threadfence();
  __syncthreads();

  // LDS byte offset of Rlds: low 32 bits of the generic pointer (LDS aperture)
  const unsigned rlds_off = (unsigned)(uintptr_t)(&Rlds[0]);

  for (int i = NSITES - 1; i >= 0; --i) {
    const float* Wm = Tbuf + (size_t)i * TWO_CHI * CHI;
    const float* Rp = Renv + (size_t)(i + 1) * CHI * CHI;
    float* Rn = Renv + (size_t)i * CHI * CHI;
    if (i > 0)
      __builtin_prefetch(Tbuf + (size_t)(i - 1) * TWO_CHI * CHI, 0, 3);

    // ---- stage R_{i+1} (128x128 f32) into LDS ----
#if HAVE_TDM
    if (wv == 0) {
      tdm_load_2d_f32(rlds_off, Rp, CHI, CHI, CHI);
      __builtin_amdgcn_s_wait_tensorcnt(0);
    }
    __syncthreads();
#else
    (void)rlds_off;
    for (int idx = tid; idx < CHI * CHI; idx += 256) Rlds[idx] = Rp[idx];
    __syncthreads();
#endif

    // GEMM1: H(256x128) = Wm(256x128) @ Rlds(128x128)
    for (int t = wv; t < 128; t += 8) {
      const int tm = t >> 3, tn = t & 7;
      v8f acc = ZERO8;
      for (int k0 = 0; k0 < CHI; k0 += 4) {
        const int ka = k0 + 2 * hi;
        v2f a, b;
        a.x = Wm[(size_t)(tm * NB + ln16) * CHI + ka];
        a.y = Wm[(size_t)(tm * NB + ln16) * CHI + ka + 1];
        b.x = Rlds[ka * CHI + tn * NB + ln16];
        b.y = Rlds[(ka + 1) * CHI + tn * NB + ln16];
        acc = wmma4(a, b, acc);
      }
      float* base = Hbuf + (size_t)(tm * NB) * CHI + tn * NB;
      for (int r = 0; r < 8; ++r)
        base[(size_t)(r + 8 * hi) * CHI + ln16] = acc[r];
    }
    __threadfence();
    __syncthreads();

    // GEMM2: Rn[a][c] = sum_{k=(s,d)} H[2a+s][d] * Wm[2c+s][d]   (K = 256)
    for (int t = wv; t < 64; t += 8) {
      const int tm = t >> 3, tn = t & 7;
      v8f acc = ZERO8;
      const int Ma = tm * NB + ln16;
      const int Nc = tn * NB + ln16;
      for (int k0 = 0; k0 < TWO_CHI; k0 += 4) {
        const int ka = k0 + 2 * hi, kb2 = ka + 1;
        v2f a, b;
        a.x = Hbuf[(size_t)(2 * Ma + (ka >> 7)) * CHI + (ka & 127)];
        a.y = Hbuf[(size_t)(2 * Ma + (kb2 >> 7)) * CHI + (kb2 & 127)];
        b.x = Wm[(size_t)(2 * Nc + (ka >> 7)) * CHI + (ka & 127)];
        b.y = Wm[(size_t)(2 * Nc + (kb2 >> 7)) * CHI + (kb2 & 127)];
        acc = wmma4(a, b, acc);
      }
      for (int r = 0; r < 8; ++r)
        Rn[(size_t)(tm * NB + r + 8 * hi) * CHI + tn * NB + ln16] = acc[r];
    }
    __threadfence();
    __syncthreads();
  }
}

// ---------------------------------------------------------------------------
// Kernel 4: autoregressive sampling sweep (single persistent workgroup).
// ---------------------------------------------------------------------------
__global__ void sample_kernel(const float* __restrict__ Tbuf,
                              const float* __restrict__ Renv,
                              const float* __restrict__ randu,
                              float* __restrict__ out,
                              float* __restrict__ scr) {
  __shared__ float red[256][4];
  __shared__ float rho_s[4];

  float* Lenv = scr;                           // 128x128
  float* Q1   = scr + (size_t)CHI * CHI;       // 256x128
  float* M3   = Q1 + (size_t)TWO_CHI * CHI;    // 256x128
  float* P1   = M3 + (size_t)TWO_CHI * CHI;    // 128x128

  const int tid = threadIdx.x;
  const int lane = tid & 31, wv = tid >> 5;
  const int ln16 = lane & 15, hi = lane >> 4;

  // Lenv = outer(l, l) = e0 e0^T
  for (int idx = tid; idx < CHI * CHI; idx += 256)
    Lenv[idx] = (idx == 0) ? 1.0f : 0.0f;
  __threadfence();
  __syncthreads();

  const float sc0 = rsqrtf(Renv[0]);  // Z = <L0, Rs[0]> = Rs[0][0][0]

  for (int i = 0; i < NSITES; ++i) {
    const float sc = (i == 0) ? sc0 : 1.0f;
    const float* Wm = Tbuf + (size_t)i * TWO_CHI * CHI;
    const float* Ri = Renv + (size_t)(i + 1) * CHI * CHI;
    if (i + 1 < NSITES)
      __builtin_prefetch(Tbuf + (size_t)(i + 1) * TWO_CHI * CHI, 0, 3);

    // Q1(256x128) = (sc*Wm) @ Ri      [= einsum('asc,cd->asd', T, Ri)]
    for (int t = wv; t < 128; t += 8) {
      const int tm = t >> 3, tn = t & 7;
      v8f acc = ZERO8;
      for (int k0 = 0; k0 < CHI; k0 += 4) {
        const int ka = k0 + 2 * hi;
        v2f a, b;
        a.x = sc * Wm[(size_t)(tm * NB + ln16) * CHI + ka];
        a.y = sc * Wm[(size_t)(tm * NB + ln16) * CHI + ka + 1];
        b.x = Ri[(size_t)ka * CHI + tn * NB + ln16];
        b.y = Ri[(size_t)(ka + 1) * CHI + tn * NB + ln16];
        acc = wmma4(a, b, acc);
      }
      float* base = Q1 + (size_t)(tm * NB) * CHI + tn * NB;
      for (int r = 0; r < 8; ++r)
        base[(size_t)(r + 8 * hi) * CHI + ln16] = acc[r];
    }
    __threadfence();
    __syncthreads();

    // M3[(s,d)][b] = sum_a Q1[2a+s][d] * Lenv[a][b]   (Lenv symmetric)
    for (int t = wv; t < 128; t += 8) {
      const int tm = t >> 3, tn = t & 7;
      v8f acc = ZERO8;
      const int Ma = tm * NB + ln16;  // combined (s,d): s=Ma>>7, d=Ma&127
      for (int k0 = 0; k0 < CHI; k0 += 4) {
        const int ka = k0 + 2 * hi;
        v2f a, b;
        a.x = Q1[(size_t)(2 * ka + (Ma >> 7)) * CHI + (Ma & 127)];
        a.y = Q1[(size_t)(2 * (ka + 1) + (Ma >> 7)) * CHI + (Ma & 127)];
        b.x = Lenv[(size_t)ka * CHI + tn * NB + ln16];
        b.y = Lenv[(size_t)(ka + 1) * CHI + tn * NB + ln16];
        acc = wmma4(a, b, acc);
      }
      float* base = M3 + (size_t)(tm * NB) * CHI + tn * NB;
      for (int r = 0; r < 8; ++r)
        base[(size_t)(r + 8 * hi) * CHI + ln16] = acc[r];
    }
    __threadfence();
    __syncthreads();

    // rho[s][t] = sum_{b,d} M3[(s,d)][b] * sc*Wm[2b+t][d]
    float a00 = 0.f, a01 = 0.f, a10 = 0.f, a11 = 0.f;
    for (int idx = tid; idx < CHI * CHI; idx += 256) {
      const int b = idx >> 7, d = idx & 127;
      const float m0 = M3[(size_t)d * CHI + b];
      const float m1 = M3[(size_t)(CHI + d) * CHI + b];
      const float g0 = sc * Wm[(size_t)(2 * b + 0) * CHI + d];
      const float g1 = sc * Wm[(size_t)(2 * b + 1) * CHI + d];
      a00 += m0 * g0; a01 += m0 * g1; a10 += m1 * g0; a11 += m1 * g1;
    }
    red[tid][0] = a00; red[tid][1] = a01; red[tid][2] = a10; red[tid][3] = a11;
    __syncthreads();
    for (int s = 128; s > 0; s >>= 1) {
      if (tid < s)
        for (int q = 0; q < 4; ++q) red[tid][q] += red[tid + s][q];
      __syncthreads();
    }
    if (tid == 0) {
      rho_s[0] = red[0][0]; rho_s[1] = red[0][1];
      rho_s[2] = red[0][2]; rho_s[3] = red[0][3];
    }
    __syncthreads();

    const float ps0 = rho_s[3];  // rho[1][1]
    const float ps1 = rho_s[0];  // rho[0][0]
    const float u = randu[i];
    const bool take0 = u < ps0;
    const float p = take0 ? ps0 : ps1;
    const int sidx = take0 ? 1 : 0;
    const float invp = 1.0f / p;
    if (tid == 0) {
      out[2 * i]     = take0 ? 0.0f : 1.0f;
      out[2 * i + 1] = take0 ? 1.0f : 0.0f;
    }

    // P1 = Lenv @ Msel,  Msel[k][n] = sc*Wm[2k+sidx][n]
    for (int t = wv; t < 64; t += 8) {
      const int tm = t >> 3, tn = t & 7;
      v8f acc = ZERO8;
      for (int k0 = 0; k0 < CHI; k0 += 4) {
        const int ka = k0 + 2 * hi;
        v2f a, b;
        a.x = Lenv[(size_t)(tm * NB + ln16) * CHI + ka];
        a.y = Lenv[(size_t)(tm * NB + ln16) * CHI + ka + 1];
        b.x = sc * Wm[(size_t)(2 * ka + sidx) * CHI + tn * NB + ln16];
        b.y = sc * Wm[(size_t)(2 * (ka + 1) + sidx) * CHI + tn * NB + ln16];
        acc = wmma4(a, b, acc);
      }
      float* base = P1 + (size_t)(tm * NB) * CHI + tn * NB;
      for (int r = 0; r < 8; ++r)
        base[(size_t)(r + 8 * hi) * CHI + ln16] = acc[r];
    }
    __threadfence();
    __syncthreads();

    // Lenv = invp * Msel^T @ P1
    for (int t = wv; t < 64; t += 8) {
      const int tm = t >> 3, tn = t & 7;
      v8f acc = ZERO8;
      for (int k0 = 0; k0 < CHI; k0 += 4) {
        const int ka = k0 + 2 * hi;
        v2f a, b;
        a.x = sc * Wm[(size_t)(2 * ka + sidx) * CHI + tm * NB + ln16];
        a.y = sc * Wm[(size_t)(2 * (ka + 1) + sidx) * CHI + tm * NB + ln16];
        b.x = P1[(size_t)ka * CHI + tn * NB + ln16];
        b.y = P1[(size_t)(ka + 1) * CHI + tn * NB + ln16];
        acc = wmma4(a, b, acc);
      }
      for (int r = 0; r < 8; ++r)
        Lenv[(size_t)(tm * NB + r + 8 * hi) * CHI + tn * NB + ln16] =
            acc[r] * invp;
    }
    __threadfence();
    __syncthreads();
  }
}

// ---------------------------------------------------------------------------
extern "C" void kernel_launch(void* const* d_in, const int* in_sizes, int n_in,
                              void* d_out, int out_size, void* d_ws,
                              size_t ws_size, hipStream_t stream) {
  const float* params = (const float*)d_in[0];   // (1024, 256, 128) f32
  const float* randu  = (const float*)d_in[1];   // (1024,) f32
  float* ws = (float*)d_ws;

  float* T    = ws + OFF_T;
  float* Renv = ws + OFF_R;
  float* scr  = ws + OFF_S;
  float* LU   = ws + OFF_LU;

  // Phase 1: Cayley isometrization (chunked; launches serialize on stream)
  for (int s0 = 0; s0 < NSITES; s0 += CHUNK) {
    build_aug_kernel<<<CHUNK, 256, 0, stream>>>(params, LU, s0);
    lu_solve_kernel<<<CHUNK, 256, 0, stream>>>(LU, T, s0);
  }
  // Phase 2: sequential right-environment scan (Hbuf aliases Q1 region)
  right_env_kernel<<<1, 256, 0, stream>>>(T, Renv, scr + (size_t)CHI * CHI);
  // Phase 3: autoregressive sampling sweep
  sample_kernel<<<1, 256, 0, stream>>>(T, Renv, randu, (float*)d_out, scr);
}